// MCT_32109175505229
// MI455X (gfx1250) — compile-verified
//
#include <hip/hip_runtime.h>
#include <hip/hip_bf16.h>
#include <math.h>

// ---------------------------------------------------------------------------
// MI455X (gfx1250) implementation.
//   N=1024, D=128, LAMB=0.5, M=N-1=1023
//   Pipeline:
//     k_gemm_g : G = f @ W1^T + b1          (fp32 WMMA 16x16x4)
//     k_magq   : smag[i]=1/|f_i|^2, q[i]=sum f_i^2 * Ws
//     k_rows   : per 16-row block: S tile via WMMA -> A=sigmoid(q(S-1)^2+bs)
//                in LDS -> row softmax (diag masked) -> T ; row0 -> p2g
//     k_copy   : x0 = p2g
//     k_matvec : x <- p2g + 0.5 * T x   (48 Richardson/Neumann steps,
//                contraction 0.5/step since T row-stochastic)
//     k_final  : out = clip(0.5 * x, 0, 1)
// ---------------------------------------------------------------------------

#define NN   1024
#define DD   128
#define MM   1023
#define ITERS 48

typedef __attribute__((ext_vector_type(2))) float v2f;
typedef __attribute__((ext_vector_type(8))) float v8f;

__device__ __forceinline__ float wave_sum(float v) {
#pragma unroll
    for (int o = 16; o > 0; o >>= 1) v += __shfl_xor(v, o, 32);
    return v;
}
__device__ __forceinline__ float wave_max(float v) {
#pragma unroll
    for (int o = 16; o > 0; o >>= 1) v = fmaxf(v, __shfl_xor(v, o, 32));
    return v;
}

// --------------------------- G = f @ W1^T + b1 ------------------------------
// 512 16x16 output tiles, one wave per tile, 8 waves/block, 64 blocks.
__global__ void __launch_bounds__(256) k_gemm_g(const float* __restrict__ f,
                                                const float* __restrict__ W1,
                                                const float* __restrict__ b1,
                                                float* __restrict__ G) {
    const int lane = threadIdx.x & 31;
    const int wave = threadIdx.x >> 5;
    const int tile = blockIdx.x * 8 + wave;          // 0..511
    const int tileM = tile >> 3;                     // 0..63  (rows of f)
    const int tileN = tile & 7;                      // 0..7   (cols of G)
    const int h   = lane >> 4;                       // half-wave select
    const int l15 = lane & 15;
    const int rowA = tileM * 16 + l15;
    const int colB = tileN * 16 + l15;

    v8f c = {};
#pragma unroll
    for (int k = 0; k < DD; k += 4) {
        const int kk = k + (h << 1);
        v2f a = *(const v2f*)(f  + rowA * DD + kk);  // A[m, kk..kk+1]
        v2f b = *(const v2f*)(W1 + colB * DD + kk);  // B[kk..kk+1, n] = W1[n, kk]
        c = __builtin_amdgcn_wmma_f32_16x16x4_f32(false, a, false, b,
                                                  (short)0, c, false, false);
    }
    const float bias = b1[colB];
#pragma unroll
    for (int r = 0; r < 8; ++r) {
        const int row = tileM * 16 + r + (h << 3);
        G[row * DD + colB] = c[r] + bias;
    }
}

// --------------------- smag = 1/|f_i|^2 , q = sum f^2 Ws --------------------
__global__ void __launch_bounds__(256) k_magq(const float* __restrict__ f,
                                              const float* __restrict__ Ws,
                                              float* __restrict__ smag,
                                              float* __restrict__ q) {
    const int lane = threadIdx.x & 31;
    const int row  = blockIdx.x * 8 + (threadIdx.x >> 5);
    float s = 0.f, qv = 0.f;
#pragma unroll
    for (int d = lane; d < DD; d += 32) {
        const float v = f[row * DD + d];
        s  += v * v;
        qv += v * v * Ws[d];
    }
    s  = wave_sum(s);
    qv = wave_sum(qv);
    if (lane == 0) { smag[row] = 1.0f / s; q[row] = qv; }
}

// ---- Per 16-row block: S via WMMA -> sigmoid -> softmax -> T (and p2g) -----
// 64 blocks x 256 threads. Wave w computes column tiles w, w+8, ... (8 tiles).
__global__ void __launch_bounds__(256) k_rows(const float* __restrict__ f,
                                              const float* __restrict__ G,
                                              const float* __restrict__ smag,
                                              const float* __restrict__ q,
                                              const float* __restrict__ bs,
                                              float* __restrict__ T,
                                              float* __restrict__ p2g) {
    __shared__ float As[16][NN];                     // 64 KB of the 320 KB WGP LDS

    const int lane = threadIdx.x & 31;
    const int wave = threadIdx.x >> 5;
    const int h    = lane >> 4;
    const int l15  = lane & 15;
    const int rowBase = blockIdx.x * 16;
    const int rowA = rowBase + l15;
    const float bsv = bs[0];

    // Register-block the A panel: f[rowBase..+15, 0..127] fragments, loaded once,
    // reused across all 8 column tiles of this wave.
    v2f afrag[DD / 4];
#pragma unroll
    for (int k4 = 0; k4 < DD / 4; ++k4)
        afrag[k4] = *(const v2f*)(f + rowA * DD + k4 * 4 + (h << 1));

    for (int i = 0; i < 8; ++i) {
        const int ct   = wave + (i << 3);            // column tile 0..63
        const int colB = ct * 16 + l15;
        v8f c = {};
#pragma unroll
        for (int k4 = 0; k4 < DD / 4; ++k4) {
            v2f b = *(const v2f*)(G + colB * DD + k4 * 4 + (h << 1));
            c = __builtin_amdgcn_wmma_f32_16x16x4_f32(false, afrag[k4], false, b,
                                                      (short)0, c, false, false);
        }
#pragma unroll
        for (int r = 0; r < 8; ++r) {
            const int rl   = r + (h << 3);
            const int grow = rowBase + rl;
            const float sv = c[r] * smag[grow];      // S[grow, colB]
            const float t  = sv - 1.0f;
            const float z  = q[grow] * t * t + bsv;
            As[rl][colB] = 1.0f / (1.0f + __expf(-z));   // A[grow, colB]
        }
    }
    __syncthreads();

    // Row softmax over j=1..1023 with diagonal masked; row 0 is p2g.
#pragma unroll
    for (int rr = 0; rr < 2; ++rr) {
        const int rl   = wave + (rr << 3);
        const int grow = rowBase + rl;
        if (grow == 0) {
            for (int j = 1 + lane; j < NN; j += 32) p2g[j - 1] = As[0][j];
        } else {
            float m = -1e30f;
            for (int j = 1 + lane; j < NN; j += 32)
                if (j != grow) m = fmaxf(m, As[rl][j]);
            m = wave_max(m);
            float s = 0.f;
            for (int j = 1 + lane; j < NN; j += 32)
                if (j != grow) s += __expf(As[rl][j] - m);
            s = wave_sum(s);
            const float inv = 1.0f / s;
            for (int j = 1 + lane; j < NN; j += 32) {
                const float tv = (j != grow) ? __expf(As[rl][j] - m) * inv : 0.0f;
                T[(grow - 1) * NN + (j - 1)] = tv;   // row stride NN (padded)
            }
        }
    }
}

// ------------------------------- x0 = p2g -----------------------------------
__global__ void k_copy(const float* __restrict__ p2g, float* __restrict__ x) {
    const int i = blockIdx.x * blockDim.x + threadIdx.x;
    if (i < MM) x[i] = p2g[i];
}

// --------------------- x_out = p2g + 0.5 * T @ x_in -------------------------
__global__ void __launch_bounds__(256) k_matvec(const float* __restrict__ T,
                                                const float* __restrict__ p2g,
                                                const float* __restrict__ xin,
                                                float* __restrict__ xout) {
    const int lane = threadIdx.x & 31;
    const int row  = blockIdx.x * 8 + (threadIdx.x >> 5);
    if (row >= MM) return;
    float s = 0.f;
    for (int j = lane; j < MM; j += 32)
        s += T[row * NN + j] * xin[j];
    s = wave_sum(s);
    if (lane == 0) xout[row] = p2g[row] + 0.5f * s;
}

// ---------------------- out = clip(0.5 * x, 0, 1) ---------------------------
__global__ void k_final(const float* __restrict__ x, float* __restrict__ out) {
    const int i = blockIdx.x * blockDim.x + threadIdx.x;
    if (i < MM) out[i] = fminf(fmaxf(0.5f * x[i], 0.0f), 1.0f);
}

// ---------------------------------------------------------------------------
extern "C" void kernel_launch(void* const* d_in, const int* in_sizes, int n_in,
                              void* d_out, int out_size, void* d_ws, size_t ws_size,
                              hipStream_t stream) {
    const float* f  = (const float*)d_in[0];   // (1024,128)
    const float* W1 = (const float*)d_in[1];   // (128,128)
    const float* b1 = (const float*)d_in[2];   // (128,)
    const float* Ws = (const float*)d_in[3];   // (1,128)
    const float* bs = (const float*)d_in[4];   // (1,)
    float* out = (float*)d_out;                // (1023,)

    float* ws   = (float*)d_ws;
    float* G    = ws;                          // 1024*128
    float* smag = G    + NN * DD;              // 1024
    float* q    = smag + NN;                   // 1024
    float* p2g  = q    + NN;                   // 1023 (pad 1024)
    float* x0   = p2g  + NN;                   // 1024
    float* x1   = x0   + NN;                   // 1024
    float* T    = x1   + NN;                   // 1023*1024 (row stride 1024)

    k_gemm_g<<<64, 256, 0, stream>>>(f, W1, b1, G);
    k_magq  <<<NN / 8, 256, 0, stream>>>(f, Ws, smag, q);
    k_rows  <<<NN / 16, 256, 0, stream>>>(f, G, smag, q, bs, T, p2g);
    k_copy  <<<(MM + 255) / 256, 256, 0, stream>>>(p2g, x0);
    for (int it = 0; it < ITERS; ++it) {
        const float* xin = (it & 1) ? x1 : x0;
        float*      xout = (it & 1) ? x0 : x1;
        k_matvec<<<NN / 8, 256, 0, stream>>>(T, p2g, xin, xout);
    }
    // ITERS even -> final result lives in x0
    k_final<<<(MM + 255) / 256, 256, 0, stream>>>(x0, out);
}